// Course_preference_48696339202412
// MI455X (gfx1250) — compile-verified
//
#include <hip/hip_runtime.h>
#include <hip/hip_bf16.h>

typedef __attribute__((ext_vector_type(2))) float v2f;
typedef __attribute__((ext_vector_type(8))) float v8f;

#define N_ITEMS 8192
#define DIM     256

// ---------------- kernel 1: per-row squared norms -> d_ws ----------------
__global__ __launch_bounds__(256) void row_sq_kernel(const float* __restrict__ emb,
                                                     float* __restrict__ sq) {
  int row = blockIdx.x * blockDim.x + threadIdx.x;
  if (row >= N_ITEMS) return;
  const float4* p = (const float4*)(emb + (size_t)row * DIM);
  float s = 0.0f;
#pragma unroll 8
  for (int i = 0; i < DIM / 4; ++i) {
    float4 v = p[i];
    s += v.x * v.x + v.y * v.y + v.z * v.z + v.w * v.w;
  }
  sq[row] = s;
}

// min-3 insert on d2 keys; ties (equal key) -> lower index, matching the
// reference's top_k tie rule under the monotone map sim = 1/(sqrt(d2)+1).
__device__ __forceinline__ void min3_insert(float v, int ix,
                                            float& t0, int& i0,
                                            float& t1, int& i1,
                                            float& t2, int& i2) {
  bool b0 = (v < t0) || (v == t0 && ix < i0);
  bool b1 = (v < t1) || (v == t1 && ix < i1);
  bool b2 = (v < t2) || (v == t2 && ix < i2);
  if (b0)      { t2 = t1; i2 = i1; t1 = t0; i1 = i0; t0 = v; i0 = ix; }
  else if (b1) { t2 = t1; i2 = i1; t1 = v;  i1 = ix; }
  else if (b2) { t2 = v;  i2 = ix; }
}

// ---------------- kernel 2: fused GEMM(f32 WMMA) + min-3(d2) + gather ----------------
__global__ __launch_bounds__(256, 1) void course_pref_kernel(
    const float* __restrict__ emb, const int* __restrict__ interact,
    const float* __restrict__ sq, float* __restrict__ out) {
  // A tile: 16 rows x 256 k, padded stride 260 floats (conflict-free b64 reads)
  __shared__ float ldsA[16 * 260];
  __shared__ float ldsSqRow[16];
  __shared__ float ldsTile[8][2][16 * 17];  // per-wave pair of 16x16 d2-key tiles
  __shared__ float ldsCV[8][16][4];         // per-wave min-3 candidate keys
  __shared__ int   ldsCI[8][16][4];         // per-wave min-3 candidate indices

  const int tid   = threadIdx.x;
  const int wave  = tid >> 5;
  const int lane  = tid & 31;
  const int half  = lane >> 4;   // 0: lanes 0-15, 1: lanes 16-31
  const int l16   = lane & 15;
  const int rbase = blockIdx.x * 16;
  const float INF = __builtin_inff();

  // cooperative stage of the 16x256 A tile into LDS (float4, coalesced)
  for (int i = tid; i < 16 * 64; i += 256) {
    int r  = i >> 6;
    int k4 = (i & 63) << 2;
    float4 v = *(const float4*)(emb + (size_t)(rbase + r) * DIM + k4);
    *(float4*)&ldsA[r * 260 + k4] = v;
  }
  if (tid < 16) ldsSqRow[tid] = sq[rbase + tid];
  __syncthreads();

  float t0 = INF, t1 = INF, t2 = INF;
  int   i0 = 0,   i1 = 0,   i2 = 0;

  // A fragment base: lane l holds M=l16, K pair {2*half, 2*half+1} per WMMA step
  const float* Abase = &ldsA[l16 * 260 + half * 2];
  float* tile0 = &ldsTile[wave][0][0];
  float* tile1 = &ldsTile[wave][1][0];

  // each wave owns a 1024-column slab -> 32 iterations of two 16x16 tiles
  for (int t = 0; t < 32; ++t) {
    const int jbase = wave * 1024 + t * 32;
    // B fragment: lane l holds N=l16 (columns jbase+l16 / jbase+16+l16)
    const float* B0 = emb + (size_t)(jbase + l16) * DIM + half * 2;
    const float* B1 = B0 + 16 * DIM;
    // warm L0 for the following tile pair (global_prefetch_b8)
    __builtin_prefetch(B1 + 16 * DIM, 0, 1);
    __builtin_prefetch(B1 + 32 * DIM, 0, 1);

    v8f acc0 = {};
    v8f acc1 = {};
#pragma unroll 8
    for (int k = 0; k < DIM; k += 4) {
      v2f a  = *(const v2f*)(Abase + k);
      v2f b0 = *(const v2f*)(B0 + k);
      v2f b1 = *(const v2f*)(B1 + k);
      // two independent accumulation chains share the A fragment (2-way WMMA ILP)
      acc0 = __builtin_amdgcn_wmma_f32_16x16x4_f32(false, a, false, b0,
                                                   (short)0, acc0, false, false);
      acc1 = __builtin_amdgcn_wmma_f32_16x16x4_f32(false, a, false, b1,
                                                   (short)0, acc1, false, false);
    }

    // d2 = |a|^2 + |b|^2 - 2 a.b ; ranking key: excluded(self) -> +INF
    const float sqn0 = sq[jbase + l16];
    const float sqn1 = sq[jbase + 16 + l16];
#pragma unroll
    for (int r = 0; r < 8; ++r) {
      int m = r + half * 8;              // C layout: VGPR r -> M = r + 8*(lane>=16)
      float s   = ldsSqRow[m];
      float d2a = fmaf(-2.0f, acc0[r], s + sqn0);
      float d2b = fmaf(-2.0f, acc1[r], s + sqn1);
      tile0[m * 17 + l16] = (d2a < 1e-10f) ? INF : d2a;
      tile1[m * 17 + l16] = (d2b < 1e-10f) ? INF : d2b;
    }

    // wave-private LDS round trip: DS is in-order per wave, just wait for stores
    asm volatile("s_wait_dscnt 0" ::: "memory");

    // lanes 0-15 scan cols 0..7 of row l16, lanes 16-31 scan cols 8..15
#pragma unroll
    for (int c = 0; c < 8; ++c) {
      int cc = c + half * 8;
      min3_insert(tile0[l16 * 17 + cc], jbase + cc,      t0, i0, t1, i1, t2, i2);
      min3_insert(tile1[l16 * 17 + cc], jbase + 16 + cc, t0, i0, t1, i1, t2, i2);
    }
    asm volatile("" ::: "memory");  // keep next tiles' DS stores after these DS loads
  }

  // merge the two half-wave candidate lists for each row (lane l <-> lane l^16)
  float p0 = __shfl_xor(t0, 16, 32); int q0 = __shfl_xor(i0, 16, 32);
  float p1 = __shfl_xor(t1, 16, 32); int q1 = __shfl_xor(i1, 16, 32);
  float p2 = __shfl_xor(t2, 16, 32); int q2 = __shfl_xor(i2, 16, 32);
  min3_insert(p0, q0, t0, i0, t1, i1, t2, i2);
  min3_insert(p1, q1, t0, i0, t1, i1, t2, i2);
  min3_insert(p2, q2, t0, i0, t1, i1, t2, i2);

  if (half == 0) {
    ldsCV[wave][l16][0] = t0; ldsCI[wave][l16][0] = i0;
    ldsCV[wave][l16][1] = t1; ldsCI[wave][l16][1] = i1;
    ldsCV[wave][l16][2] = t2; ldsCI[wave][l16][2] = i2;
  }
  __syncthreads();

  // final 8-way merge, then exact reference math on the 3 winners only:
  // sim = 1/(sqrt(key)+1); key==INF (self/excluded) naturally yields sim = 0
  if (tid < 16) {
    float f0 = INF, f1 = INF, f2 = INF;
    int   j0 = 0,   j1 = 0,   j2 = 0;
#pragma unroll
    for (int w = 0; w < 8; ++w) {
      min3_insert(ldsCV[w][tid][0], ldsCI[w][tid][0], f0, j0, f1, j1, f2, j2);
      min3_insert(ldsCV[w][tid][1], ldsCI[w][tid][1], f0, j0, f1, j1, f2, j2);
      min3_insert(ldsCV[w][tid][2], ldsCI[w][tid][2], f0, j0, f1, j1, f2, j2);
    }
    float s0 = 1.0f / (sqrtf(f0) + 1.0f);
    float s1 = 1.0f / (sqrtf(f1) + 1.0f);
    float s2 = 1.0f / (sqrtf(f2) + 1.0f);
    const int rg = rbase + tid;
    const int* __restrict__ irow = interact + (size_t)rg * N_ITEMS;
    float p = s0 * (float)irow[j0] + s1 * (float)irow[j1] + s2 * (float)irow[j2];
    out[rg] = p * (1.0f / 3.0f);
  }
}

extern "C" void kernel_launch(void* const* d_in, const int* in_sizes, int n_in,
                              void* d_out, int out_size, void* d_ws, size_t ws_size,
                              hipStream_t stream) {
  const float* emb      = (const float*)d_in[0];   // [8192, 256] f32
  const int*   interact = (const int*)d_in[1];     // [8192, 8192] i32
  float*       sq       = (float*)d_ws;            // 8192 f32 scratch (32 KB)
  float*       out      = (float*)d_out;           // [8192] f32

  row_sq_kernel<<<N_ITEMS / 256, 256, 0, stream>>>(emb, sq);
  course_pref_kernel<<<N_ITEMS / 16, 256, 0, stream>>>(emb, interact, sq, out);
}